// GCN_33389075759722
// MI455X (gfx1250) — compile-verified
//
#include <hip/hip_runtime.h>

#define N_NODES  50000
#define N_EDGES  600000
#define D_FEAT   128
#define N_CLASSES 40

typedef __attribute__((ext_vector_type(2))) float v2f;
typedef __attribute__((ext_vector_type(8))) float v8f;

// ---------------------------------------------------------------------------
// agg = h   (self term; scatter adds on top of this)
// ---------------------------------------------------------------------------
__global__ void gcn_copy(const float4* __restrict__ src, float4* __restrict__ dst, int n4) {
    int i = blockIdx.x * blockDim.x + threadIdx.x;
    if (i < n4) dst[i] = src[i];
}

// ---------------------------------------------------------------------------
// agg[dst[e]] += h[src[e]]   : 32 lanes per edge, float4 per lane
// ---------------------------------------------------------------------------
__global__ void gcn_scatter(const float* __restrict__ h,
                            const int* __restrict__ src,
                            const int* __restrict__ dst,
                            float* __restrict__ agg) {
    long long gid = (long long)blockIdx.x * blockDim.x + threadIdx.x;
    int edge = (int)(gid >> 5);
    if (edge >= N_EDGES) return;
    int lane = (int)(gid & 31);
    int s = src[edge];
    int d = dst[edge];
    const float4 v = *(const float4*)(h + (size_t)s * D_FEAT + lane * 4);
    float* p = agg + (size_t)d * D_FEAT + lane * 4;
    atomicAdd(p + 0, v.x);
    atomicAdd(p + 1, v.y);
    atomicAdd(p + 2, v.z);
    atomicAdd(p + 3, v.w);
}

// ---------------------------------------------------------------------------
// out[n, j] = act( sum_k A[n,k] * W[j,k] + bias[j] )
// Block: 64 nodes (4 row tiles of 16) x full output width.
// One wave per 16-column tile; blockDim.x = 32 * num_col_tiles.
// K loop in steps of 4 using V_WMMA_F32_16X16X4_F32.
// ---------------------------------------------------------------------------
#define ROWS 64
#define LDA  132   // padded LDS pitch (floats): 528B = 33*16B -> aligned + conflict-free

template <bool RELU>
__global__ __launch_bounds__(256) void gcn_gemm(const float* __restrict__ A,
                                                const float* __restrict__ W,
                                                const float* __restrict__ bias,
                                                float* __restrict__ out,
                                                int dout) {
    __shared__ float ldsA[ROWS * LDA];

    const int tid     = threadIdx.x;
    const int wave    = tid >> 5;
    const int lane    = tid & 31;
    const int rowbase = blockIdx.x * ROWS;

    // Stage A tile (64 x 128) into LDS, zero-padding out-of-range rows.
    for (int idx = tid; idx < ROWS * (D_FEAT / 4); idx += blockDim.x) {
        int r  = idx >> 5;          // 0..63
        int c4 = idx & 31;          // 0..31 (float4 columns)
        int node = rowbase + r;
        float4 v = make_float4(0.f, 0.f, 0.f, 0.f);
        if (node < N_NODES) v = *(const float4*)(A + (size_t)node * D_FEAT + c4 * 4);
        *(float4*)(&ldsA[r * LDA + c4 * 4]) = v;
    }
    __syncthreads();

    // WMMA fragment addressing (ISA 7.12.2):
    //  A 16x4 f32 : lanes 0-15 hold M=lane {K,K+1}; lanes 16-31 hold M=lane-16 {K+2,K+3}
    //  B 4x16 f32 : lanes 0-15 hold N=lane rows {K,K+1}; lanes 16-31 rows {K+2,K+3}
    const int m    = lane & 15;
    const int koff = (lane >> 4) << 1;
    const int j    = wave * 16 + m;                    // output column this lane feeds B for
    const int jc   = (j < dout) ? j : (dout - 1);      // clamp: no divergence around WMMA
    const float* wrow = W + (size_t)jc * D_FEAT;

    v8f acc0 = {0.f,0.f,0.f,0.f,0.f,0.f,0.f,0.f};
    v8f acc1 = acc0, acc2 = acc0, acc3 = acc0;

    for (int k = 0; k < D_FEAT; k += 4) {
        v2f b = *(const v2f*)(wrow + k + koff);
        v2f a0 = *(const v2f*)(&ldsA[( 0 + m) * LDA + k + koff]);
        v2f a1 = *(const v2f*)(&ldsA[(16 + m) * LDA + k + koff]);
        v2f a2 = *(const v2f*)(&ldsA[(32 + m) * LDA + k + koff]);
        v2f a3 = *(const v2f*)(&ldsA[(48 + m) * LDA + k + koff]);
        acc0 = __builtin_amdgcn_wmma_f32_16x16x4_f32(false, a0, false, b, (short)0, acc0, false, false);
        acc1 = __builtin_amdgcn_wmma_f32_16x16x4_f32(false, a1, false, b, (short)0, acc1, false, false);
        acc2 = __builtin_amdgcn_wmma_f32_16x16x4_f32(false, a2, false, b, (short)0, acc2, false, false);
        acc3 = __builtin_amdgcn_wmma_f32_16x16x4_f32(false, a3, false, b, (short)0, acc3, false, false);
    }

    // C/D layout: VGPR r holds row M = r + 8*(lane>=16), col N = lane&15.
    const int half = lane >> 4;
    const int col  = wave * 16 + (lane & 15);
    float bv = (col < dout) ? bias[col] : 0.f;

    v8f accs[4] = {acc0, acc1, acc2, acc3};
    #pragma unroll
    for (int rt = 0; rt < 4; ++rt) {
        #pragma unroll
        for (int r = 0; r < 8; ++r) {
            int node = rowbase + rt * 16 + r + half * 8;
            if (node < N_NODES && col < dout) {
                float v = accs[rt][r] + bv;
                if (RELU) v = fmaxf(v, 0.f);
                out[(size_t)node * dout + col] = v;
            }
        }
    }
}

// ---------------------------------------------------------------------------
extern "C" void kernel_launch(void* const* d_in, const int* in_sizes, int n_in,
                              void* d_out, int out_size, void* d_ws, size_t ws_size,
                              hipStream_t stream) {
    const float* x   = (const float*)d_in[0];
    const int*   src = (const int*)d_in[1];
    const int*   dst = (const int*)d_in[2];
    const float* W0  = (const float*)d_in[3];
    const float* b0  = (const float*)d_in[4];
    const float* W1  = (const float*)d_in[5];
    const float* b1  = (const float*)d_in[6];
    const float* W2  = (const float*)d_in[7];
    const float* b2  = (const float*)d_in[8];
    float* out = (float*)d_out;

    const size_t HSZ = (size_t)N_NODES * D_FEAT;   // floats per feature buffer
    float* agg = (float*)d_ws;
    float* hA  = agg + HSZ;
    float* hB  = hA + HSZ;

    const int  n4 = N_NODES * (D_FEAT / 4);
    dim3 cgrid((n4 + 255) / 256), cblk(256);
    long long sthreads = (long long)N_EDGES * 32;
    dim3 sgrid((unsigned)((sthreads + 255) / 256)), sblk(256);
    dim3 ggrid((N_NODES + ROWS - 1) / ROWS);

    // Layer 0: x -> hA (relu)
    gcn_copy<<<cgrid, cblk, 0, stream>>>((const float4*)x, (float4*)agg, n4);
    gcn_scatter<<<sgrid, sblk, 0, stream>>>(x, src, dst, agg);
    gcn_gemm<true><<<ggrid, dim3(256), 0, stream>>>(agg, W0, b0, hA, D_FEAT);

    // Layer 1: hA -> hB (relu)
    gcn_copy<<<cgrid, cblk, 0, stream>>>((const float4*)hA, (float4*)agg, n4);
    gcn_scatter<<<sgrid, sblk, 0, stream>>>(hA, src, dst, agg);
    gcn_gemm<true><<<ggrid, dim3(256), 0, stream>>>(agg, W1, b1, hB, D_FEAT);

    // Layer 2: hB -> out (no act, 40 cols -> 3 column tiles -> 96 threads)
    gcn_copy<<<cgrid, cblk, 0, stream>>>((const float4*)hB, (float4*)agg, n4);
    gcn_scatter<<<sgrid, sblk, 0, stream>>>(hB, src, dst, agg);
    gcn_gemm<false><<<ggrid, dim3(96), 0, stream>>>(agg, W2, b2, out, N_CLASSES);
}